// modelRNN_58884001628723
// MI455X (gfx1250) — compile-verified
//
#include <hip/hip_runtime.h>
#include <hip/hip_bf16.h>

typedef __attribute__((ext_vector_type(2))) float v2f;
typedef __attribute__((ext_vector_type(8))) float v8f;

#define Bsz 256
#define Tt  1024
#define INf 128
#define Hh  10
#define OUTf 128
// M = Bsz*Tt = 262144 rows

// ---------------------------------------------------------------------------
// K1: xp0[t][b][0:16] = x[b,t,:] @ W_ih0^T + (b_ih0 + b_hh0), cols 10..15 = 0
// One wave per 16-row tile, f32 WMMA 16x16x4, K=128 -> 32 wmma per tile.
// ---------------------------------------------------------------------------
__global__ __launch_bounds__(256) void k_xp0(const float* __restrict__ x,
                                             const float* __restrict__ Wih0,
                                             const float* __restrict__ bih0,
                                             const float* __restrict__ bhh0,
                                             float* __restrict__ xp0) {
  // B matrix (K=128 x N=16, zero-padded cols>=10) stored as K-pairs:
  // wp[p*16+n] = { W[n][2p], W[n][2p+1] }  (i.e. {B[2p][n], B[2p+1][n]})
  __shared__ v2f  wp[64 * 16];
  __shared__ float bsum[16];

  const int tid = threadIdx.x;
  for (int i = tid; i < 64 * 16; i += 256) {
    int p = i >> 4, n = i & 15;
    v2f w; w.x = 0.f; w.y = 0.f;
    if (n < Hh) { w.x = Wih0[n * INf + 2 * p]; w.y = Wih0[n * INf + 2 * p + 1]; }
    wp[i] = w;
  }
  if (tid < 16) bsum[tid] = (tid < Hh) ? (bih0[tid] + bhh0[tid]) : 0.f;
  __syncthreads();

  const int wave = tid >> 5, lane = tid & 31;
  const int n = lane & 15, hi = lane >> 4;
  const long long m0 = ((long long)blockIdx.x * 8 + wave) * 16;

  const float* arow = x + (m0 + n) * (long long)INf + 2 * hi; // A: m=m0+(l&15)

  v8f acc;
  const float bv = bsum[n];
#pragma unroll
  for (int i = 0; i < 8; i++) acc[i] = bv;

#pragma unroll
  for (int k0 = 0; k0 < INf; k0 += 4) {
    v2f a = *(const v2f*)(arow + k0);                // {x[m][k0+2hi], x[m][k0+2hi+1]}
    v2f b = wp[((k0 >> 1) + hi) * 16 + n];           // {B[k0+2hi][n], B[k0+2hi+1][n]}
    acc = __builtin_amdgcn_wmma_f32_16x16x4_f32(false, a, false, b,
                                                (short)0, acc, false, false);
  }

#pragma unroll
  for (int r = 0; r < 8; r++) {
    long long mg = m0 + r + 8 * hi;                  // global row b*T + t
    int t  = (int)(mg & (Tt - 1));
    int bb = (int)(mg >> 10);
    xp0[((long long)t * Bsz + bb) * 16 + n] = acc[r];
  }
}

// ---------------------------------------------------------------------------
// K2: fused 2-layer recurrent scan. One lane per batch element (8 waves total).
// All weights register-resident; per step: 10x10 + 10x10 + 10x10 fused matvecs.
// ---------------------------------------------------------------------------
__global__ __launch_bounds__(32) void k_scan(const float* __restrict__ xp0,
                                             const float* __restrict__ Whh0,
                                             const float* __restrict__ Wih1,
                                             const float* __restrict__ Whh1,
                                             const float* __restrict__ bih1,
                                             const float* __restrict__ bhh1,
                                             float* __restrict__ out2,
                                             float* __restrict__ hidden) {
  const int b = blockIdx.x * 32 + threadIdx.x;

  float w0[100], wi[100], w1[100], bb1[10];
#pragma unroll
  for (int i = 0; i < 100; i++) { w0[i] = Whh0[i]; wi[i] = Wih1[i]; w1[i] = Whh1[i]; }
#pragma unroll
  for (int i = 0; i < 10; i++) bb1[i] = bih1[i] + bhh1[i];

  float h0[10], h1[10];
#pragma unroll
  for (int i = 0; i < 10; i++) { h0[i] = 0.f; h1[i] = 0.f; }

  for (int t = 0; t < Tt; t++) {
    const long long row = (long long)t * Bsz + b;
    const float4* xr = (const float4*)(xp0 + row * 16);
    float4 p0 = xr[0], p1 = xr[1];
    float2 p2 = *(const float2*)(xp0 + row * 16 + 8);
    float xv[10] = {p0.x, p0.y, p0.z, p0.w, p1.x, p1.y, p1.z, p1.w, p2.x, p2.y};

    float nh0[10];
#pragma unroll
    for (int o = 0; o < 10; o++) {
      float s = xv[o];
#pragma unroll
      for (int j = 0; j < 10; j++) s = fmaf(h0[j], w0[o * 10 + j], s);
      nh0[o] = fmaxf(s, 0.f);
    }
    float nh1[10];
#pragma unroll
    for (int o = 0; o < 10; o++) {
      float s = bb1[o];
#pragma unroll
      for (int j = 0; j < 10; j++) s = fmaf(nh0[j], wi[o * 10 + j], s);
#pragma unroll
      for (int j = 0; j < 10; j++) s = fmaf(h1[j], w1[o * 10 + j], s);
      nh1[o] = fmaxf(s, 0.f);
    }
#pragma unroll
    for (int i = 0; i < 10; i++) { h0[i] = nh0[i]; h1[i] = nh1[i]; }

    float4* orow = (float4*)(out2 + row * 16);
    orow[0] = make_float4(h1[0], h1[1], h1[2], h1[3]);
    orow[1] = make_float4(h1[4], h1[5], h1[6], h1[7]);
    orow[2] = make_float4(h1[8], h1[9], 0.f, 0.f);
    orow[3] = make_float4(0.f, 0.f, 0.f, 0.f);
  }

  // hidden: [2][B][H]
#pragma unroll
  for (int i = 0; i < 10; i++) {
    hidden[b * Hh + i]              = h0[i];
    hidden[Bsz * Hh + b * Hh + i]   = h1[i];
  }
}

// ---------------------------------------------------------------------------
// K3: logits = out2 @ W_lin^T + b_lin via WMMA (K=16 padded), fused softmax.
// One wave per 16-row tile; 8 column tiles x 4 k-steps = 32 wmma per tile.
// ---------------------------------------------------------------------------
__global__ __launch_bounds__(256) void k_out(const float* __restrict__ out2,
                                             const float* __restrict__ Wlin,
                                             const float* __restrict__ blin,
                                             float* __restrict__ out) {
  __shared__ v2f  wp[8 * 128];  // wp[p*128+o] = {Wlin[o][2p], Wlin[o][2p+1]} (0-pad k>=10)
  __shared__ float bl[128];

  const int tid = threadIdx.x;
  for (int i = tid; i < 8 * 128; i += 256) {
    int p = i >> 7, o = i & 127;
    v2f w;
    w.x = (2 * p     < Hh) ? Wlin[o * Hh + 2 * p]     : 0.f;
    w.y = (2 * p + 1 < Hh) ? Wlin[o * Hh + 2 * p + 1] : 0.f;
    wp[i] = w;
  }
  if (tid < 128) bl[tid] = blin[tid];
  __syncthreads();

  const int wave = tid >> 5, lane = tid & 31;
  const int n = lane & 15, hi = lane >> 4;
  const long long m0 = ((long long)blockIdx.x * 8 + wave) * 16;

  // A row for this lane: logits row m0+n lives at storage row t*B+b
  const long long am = m0 + n;
  const int at = (int)(am & (Tt - 1));
  const int ab = (int)(am >> 10);
  const float* abase = out2 + ((long long)at * Bsz + ab) * 16 + 2 * hi;
  v2f a0 = *(const v2f*)(abase + 0);
  v2f a1 = *(const v2f*)(abase + 4);
  v2f a2 = *(const v2f*)(abase + 8);
  v2f a3 = *(const v2f*)(abase + 12);

  v8f acc[8];
#pragma unroll
  for (int ct = 0; ct < 8; ct++) {
    const int o = ct * 16 + n;
    const float bv = bl[o];
#pragma unroll
    for (int i = 0; i < 8; i++) acc[ct][i] = bv;
    v2f b0 = wp[(0 + hi) * 128 + o];
    v2f b1 = wp[(2 + hi) * 128 + o];
    v2f b2 = wp[(4 + hi) * 128 + o];
    v2f b3 = wp[(6 + hi) * 128 + o];
    acc[ct] = __builtin_amdgcn_wmma_f32_16x16x4_f32(false, a0, false, b0, (short)0, acc[ct], false, false);
    acc[ct] = __builtin_amdgcn_wmma_f32_16x16x4_f32(false, a1, false, b1, (short)0, acc[ct], false, false);
    acc[ct] = __builtin_amdgcn_wmma_f32_16x16x4_f32(false, a2, false, b2, (short)0, acc[ct], false, false);
    acc[ct] = __builtin_amdgcn_wmma_f32_16x16x4_f32(false, a3, false, b3, (short)0, acc[ct], false, false);
  }

  // Fused softmax: VGPR r holds rows {m0+r (lanes 0-15), m0+r+8 (lanes 16-31)},
  // cols = ct*16 + (lane&15). Reduce over 8 ct locally, then 16-lane butterfly.
#pragma unroll
  for (int r = 0; r < 8; r++) {
    float mx = acc[0][r];
#pragma unroll
    for (int ct = 1; ct < 8; ct++) mx = fmaxf(mx, acc[ct][r]);
#pragma unroll
    for (int off = 1; off < 16; off <<= 1) mx = fmaxf(mx, __shfl_xor(mx, off, 16));

    float e[8];
    float s = 0.f;
#pragma unroll
    for (int ct = 0; ct < 8; ct++) { e[ct] = __expf(acc[ct][r] - mx); s += e[ct]; }
#pragma unroll
    for (int off = 1; off < 16; off <<= 1) s += __shfl_xor(s, off, 16);
    const float inv = 1.0f / s;

    const long long rowm = m0 + r + 8 * hi;
    float* op = out + rowm * (long long)OUTf + n;
#pragma unroll
    for (int ct = 0; ct < 8; ct++) op[ct * 16] = e[ct] * inv;
  }
}

// ---------------------------------------------------------------------------
extern "C" void kernel_launch(void* const* d_in, const int* in_sizes, int n_in,
                              void* d_out, int out_size, void* d_ws, size_t ws_size,
                              hipStream_t stream) {
  const float* x    = (const float*)d_in[0];
  const float* Wih0 = (const float*)d_in[1];
  const float* Whh0 = (const float*)d_in[2];
  const float* bih0 = (const float*)d_in[3];
  const float* bhh0 = (const float*)d_in[4];
  const float* Wih1 = (const float*)d_in[5];
  const float* Whh1 = (const float*)d_in[6];
  const float* bih1 = (const float*)d_in[7];
  const float* bhh1 = (const float*)d_in[8];
  const float* Wlin = (const float*)d_in[9];
  const float* blin = (const float*)d_in[10];

  float* out    = (float*)d_out;                                  // [B,T,128] softmax
  float* hidden = out + (long long)Bsz * Tt * OUTf;               // [2,B,10]

  float* xp0  = (float*)d_ws;                                     // [T][B][16] f32
  float* out2 = xp0 + (long long)Tt * Bsz * 16;                   // [T][B][16] f32

  // 262144 rows / 16 rows per wave / 8 waves per block = 2048 blocks
  k_xp0 <<<2048, 256, 0, stream>>>(x, Wih0, bih0, bhh0, xp0);
  k_scan<<<8,    32,  0, stream>>>(xp0, Whh0, Wih1, Whh1, bih1, bhh1, out2, hidden);
  k_out <<<2048, 256, 0, stream>>>(out2, Wlin, blin, out);
}